// GNNNet_83958020703045
// MI455X (gfx1250) — compile-verified
//
#include <hip/hip_runtime.h>
#include <cstdint>
#include <cstddef>

typedef __attribute__((ext_vector_type(2))) float v2f;
typedef __attribute__((ext_vector_type(8))) float v8f;

// ------------------------------------------------------------------
// utility: zero a float buffer (harness poisons d_out/d_ws with 0xAA)
// ------------------------------------------------------------------
__global__ void k_zero(float* __restrict__ p, size_t n) {
  size_t i = (size_t)blockIdx.x * blockDim.x + threadIdx.x;
  size_t stride = (size_t)gridDim.x * blockDim.x;
  for (; i < n; i += stride) p[i] = 0.0f;
}

// ------------------------------------------------------------------
// degree count over edge destinations (self-loop handled in k_dinv)
// ------------------------------------------------------------------
__global__ void k_degree(const long long* __restrict__ dstI, float* __restrict__ deg, int E) {
  int e = blockIdx.x * blockDim.x + threadIdx.x;
  if (e < E) atomicAdd(&deg[dstI[e]], 1.0f);
}

__global__ void k_dinv(const float* __restrict__ deg, float* __restrict__ dinv, int N) {
  int i = blockIdx.x * blockDim.x + threadIdx.x;
  if (i < N) dinv[i] = rsqrtf(deg[i] + 1.0f);  // +1 self-loop => deg >= 1 always
}

// ------------------------------------------------------------------
// fp32 WMMA GEMM: H[N,NOUT] = X[N,K] @ W[K,NOUT]
// One wave computes 16 rows x NOUT cols using V_WMMA_F32_16X16X4_F32.
// K and NOUT are compile-time so the N-tile loop fully unrolls and the
// accumulators stay in VGPRs (no scratch spill).
// A 16x4 f32 fragment layout (ISA 7.12.2):
//   lanes 0-15 : v[0]=A[m][k+0], v[1]=A[m][k+1]
//   lanes 16-31: v[0]=A[m][k+2], v[1]=A[m][k+3]
// B 4x16 fragment is the symmetric (K-major) layout.
// C/D 16x16 f32: lanes 0-15 -> rows 0..7, lanes 16-31 -> rows 8..15.
// Requires N % 16 == 0 (50000/16 = 3125, exact).
// ------------------------------------------------------------------
template <int K, int NOUT>
__global__ __launch_bounds__(256)
void k_gemm_wmma_f32(const float* __restrict__ X, const float* __restrict__ W,
                     float* __restrict__ H, int nTilesM) {
  constexpr int NT = NOUT / 16;         // 8 or 4 accumulator tiles
  const int wavesPerBlock = blockDim.x >> 5;
  const int wave = blockIdx.x * wavesPerBlock + (threadIdx.x >> 5);
  if (wave >= nTilesM) return;          // wave-uniform guard: EXEC all-ones for WMMA
  const int lane = threadIdx.x & 31;
  const int half = lane >> 4;           // 0 => lanes 0-15, 1 => lanes 16-31
  const int l16  = lane & 15;
  const int mbase = wave << 4;

  const v8f vzero = {0.f, 0.f, 0.f, 0.f, 0.f, 0.f, 0.f, 0.f};
  v8f acc[NT];
#pragma unroll
  for (int t = 0; t < NT; ++t) acc[t] = vzero;

  // A-fragment pointer: row (mbase+l16), columns k + 2*half .. +1 (8B aligned)
  const float* xrow = X + (size_t)(mbase + l16) * K + (half << 1);
  for (int k = 0; k < K; k += 4) {
    const v2f a = *reinterpret_cast<const v2f*>(xrow + k);
    const float* wrow = W + (size_t)(k + (half << 1)) * NOUT + l16;
#pragma unroll
    for (int t = 0; t < NT; ++t) {
      v2f b;
      b.x = wrow[t * 16];
      b.y = wrow[t * 16 + NOUT];
      acc[t] = __builtin_amdgcn_wmma_f32_16x16x4_f32(
          /*neg_a=*/false, a, /*neg_b=*/false, b,
          /*c_mod=*/(short)0, acc[t], /*reuse_a=*/false, /*reuse_b=*/false);
    }
  }

  const int rbase = mbase + (half << 3);
#pragma unroll
  for (int t = 0; t < NT; ++t)
#pragma unroll
    for (int v = 0; v < 8; ++v)
      H[(size_t)(rbase + v) * NOUT + t * 16 + l16] = acc[t][v];
}

// ------------------------------------------------------------------
// edge scatter: agg[dst] += H[src] * dinv[src]*dinv[dst]
// one thread per (edge, 4-feature chunk); D is a power of two
// ------------------------------------------------------------------
__global__ void k_scatter(const float* __restrict__ H, const long long* __restrict__ srcI,
                          const long long* __restrict__ dstI, const float* __restrict__ dinv,
                          float* __restrict__ agg, int E, int d4shift, int D) {
  const int D4 = D >> 2;
  const int epb = blockDim.x >> d4shift;                      // edges per block
  const int e = blockIdx.x * epb + (threadIdx.x >> d4shift);
  const int c = threadIdx.x & (D4 - 1);
  if (e >= E) return;
  const long long s = srcI[e];
  const long long d = dstI[e];
  const float w = dinv[s] * dinv[d];
  const float4 hv = reinterpret_cast<const float4*>(H + (size_t)s * D)[c];
  float* o = agg + (size_t)d * D + ((size_t)c << 2);
  atomicAdd(o + 0, hv.x * w);
  atomicAdd(o + 1, hv.y * w);
  atomicAdd(o + 2, hv.z * w);
  atomicAdd(o + 3, hv.w * w);
}

// ------------------------------------------------------------------
// finish: agg = act(agg + H*dinv^2 (self-loop) + bias), in place
// ------------------------------------------------------------------
__global__ void k_finish(const float* __restrict__ H, const float* __restrict__ dinv,
                         const float* __restrict__ bias, float* __restrict__ agg,
                         size_t total, int dshift, int D, int doRelu) {
  size_t i = (size_t)blockIdx.x * blockDim.x + threadIdx.x;
  if (i >= total) return;
  const int n = (int)(i >> dshift);
  const int f = (int)(i & (size_t)(D - 1));
  const float di = dinv[n];
  float v = agg[i] + H[i] * di * di + bias[f];
  agg[i] = doRelu ? fmaxf(v, 0.0f) : v;
}

// ------------------------------------------------------------------
// launch
// ------------------------------------------------------------------
extern "C" void kernel_launch(void* const* d_in, const int* in_sizes, int n_in,
                              void* d_out, int out_size, void* d_ws, size_t ws_size,
                              hipStream_t stream) {
  const float*     x   = (const float*)d_in[0];
  const long long* ei  = (const long long*)d_in[1];   // int64 [2, E]
  const float*     W1  = (const float*)d_in[2];
  const float*     b1  = (const float*)d_in[3];
  const float*     W2  = (const float*)d_in[4];
  const float*     b2  = (const float*)d_in[5];
  const float*     W3  = (const float*)d_in[6];
  const float*     b3  = (const float*)d_in[7];
  float* out = (float*)d_out;
  (void)n_in; (void)out_size; (void)ws_size;

  const int IN = 128, HID = 128, EMB = 64;
  const int N = in_sizes[0] / IN;   // 50000
  const int E = in_sizes[1] / 2;    // 800000
  const long long* srcI = ei;
  const long long* dstI = ei + E;

  // workspace layout (256B aligned slices)
  char* ws = (char*)d_ws;
  size_t off = 0;
  auto take = [&](size_t bytes) -> float* {
    float* p = (float*)(ws + off);
    off = (off + bytes + 255) & ~(size_t)255;
    return p;
  };
  float* deg  = take((size_t)N * sizeof(float));
  float* dinv = take((size_t)N * sizeof(float));
  float* bufH = take((size_t)N * 128 * sizeof(float));   // transformed features
  float* bufA = take((size_t)N * 128 * sizeof(float));   // aggregation / activations

  const int T = 256;
  auto cdiv = [](long long a, long long b) { return (int)((a + b - 1) / b); };

  // --- symmetric normalization weights ---
  k_zero  <<<cdiv(N, T), T, 0, stream>>>(deg, (size_t)N);
  k_degree<<<cdiv(E, T), T, 0, stream>>>(dstI, deg, E);
  k_dinv  <<<cdiv(N, T), T, 0, stream>>>(deg, dinv, N);

  const int tilesM = N / 16;                       // 3125
  const int gemmBlocks = cdiv(tilesM, T / 32);     // 8 waves per block

  // --- layer 1: 128 -> 128, ReLU ---
  k_gemm_wmma_f32<128, 128><<<gemmBlocks, T, 0, stream>>>(x, W1, bufH, tilesM);
  k_zero   <<<cdiv((long long)N * HID, T), T, 0, stream>>>(bufA, (size_t)N * HID);
  k_scatter<<<cdiv((long long)E * (HID / 4), T), T, 0, stream>>>(bufH, srcI, dstI, dinv, bufA, E, 5, HID);
  k_finish <<<cdiv((long long)N * HID, T), T, 0, stream>>>(bufH, dinv, b1, bufA, (size_t)N * HID, 7, HID, 1);

  // --- layer 2: 128 -> 64, ReLU ---
  k_gemm_wmma_f32<128, 64><<<gemmBlocks, T, 0, stream>>>(bufA, W2, bufH, tilesM);
  k_zero   <<<cdiv((long long)N * EMB, T), T, 0, stream>>>(bufA, (size_t)N * EMB);
  k_scatter<<<cdiv((long long)E * (EMB / 4), T), T, 0, stream>>>(bufH, srcI, dstI, dinv, bufA, E, 4, EMB);
  k_finish <<<cdiv((long long)N * EMB, T), T, 0, stream>>>(bufH, dinv, b2, bufA, (size_t)N * EMB, 6, EMB, 1);

  // --- layer 3: 64 -> 128, no activation, into d_out ---
  k_gemm_wmma_f32<64, 128><<<gemmBlocks, T, 0, stream>>>(bufA, W3, bufH, tilesM);
  k_zero   <<<cdiv((long long)N * IN, T), T, 0, stream>>>(out, (size_t)N * IN);
  k_scatter<<<cdiv((long long)E * (IN / 4), T), T, 0, stream>>>(bufH, srcI, dstI, dinv, out, E, 5, IN);
  k_finish <<<cdiv((long long)N * IN, T), T, 0, stream>>>(bufH, dinv, b3, out, (size_t)N * IN, 7, IN, 0);
}